// EmbLoss_v2_4776003633319
// MI455X (gfx1250) — compile-verified
//
#include <hip/hip_runtime.h>
#include <math.h>

#define FEATURE_DIM 4
#define KNUM 32
#define BG_SAMPLE 100
#define CHUNK 8192
#define DELTA_V 0.5f
#define DELTA_D 1.5f

typedef float v2f __attribute__((ext_vector_type(2)));
typedef float v8f __attribute__((ext_vector_type(8)));

// ---------------- zero scratch accumulators ----------------
__global__ void k_zero(float* p, int n) {
    int i = blockIdx.x * blockDim.x + threadIdx.x;
    if (i < n) p[i] = 0.0f;
}

// ---------------- pass 1: segment counts/sums + bg counts per chunk ---------
__global__ void k_stats(const float* __restrict__ emb, const int* __restrict__ inst,
                        const float* __restrict__ ker, const float* __restrict__ tmask,
                        float* cnt_k, float* sums, float* cnt_all, int* bgcnt,
                        int N, int nblk) {
    int b   = blockIdx.x / nblk;
    int blk = blockIdx.x - b * nblk;
    __shared__ float s_sums[KNUM][FEATURE_DIM];
    __shared__ float s_cntk[KNUM];
    __shared__ float s_cnta[KNUM];
    __shared__ int   s_bg;
    int t = threadIdx.x;
    if (t < KNUM) {
        s_cntk[t] = 0.f; s_cnta[t] = 0.f;
        for (int d = 0; d < FEATURE_DIM; ++d) s_sums[t][d] = 0.f;
    }
    if (t == 0) s_bg = 0;
    __syncthreads();

    const float* eb = emb + (size_t)b * FEATURE_DIM * N;
    const int*   ib = inst + (size_t)b * N;
    const float* kb = ker  + (size_t)b * N;
    const float* tb = tmask + (size_t)b * N;
    int beg = blk * CHUNK;
    int end = beg + CHUNK; if (end > N) end = N;
    int mybg = 0;
    for (int i = beg + t; i < end; i += blockDim.x) {
        int lab = ib[i];
        int l  = (tb[i] > 0.5f) ? lab : 0;   // inst2d
        int lk = (kb[i] > 0.5f) ? l   : 0;   // kernel-restricted
        atomicAdd(&s_cnta[l], 1.0f);
        atomicAdd(&s_cntk[lk], 1.0f);
        atomicAdd(&s_sums[lk][0], eb[i]);
        atomicAdd(&s_sums[lk][1], eb[(size_t)N + i]);
        atomicAdd(&s_sums[lk][2], eb[(size_t)2 * N + i]);
        atomicAdd(&s_sums[lk][3], eb[(size_t)3 * N + i]);
        if (l == 0) mybg++;
    }
    atomicAdd(&s_bg, mybg);
    __syncthreads();
    if (t < KNUM) {
        atomicAdd(&cnt_all[b * KNUM + t], s_cnta[t]);
        atomicAdd(&cnt_k[b * KNUM + t],  s_cntk[t]);
        for (int d = 0; d < FEATURE_DIM; ++d)
            atomicAdd(&sums[(b * KNUM + t) * FEATURE_DIM + d], s_sums[t][d]);
    }
    if (t == 0) bgcnt[b * nblk + blk] = s_bg;
}

// ---------------- means / norms / exp(md) + exclusive scan of bg counts -----
__global__ void k_prep(const float* cnt_k, const float* sums, const float* md,
                       float* mean, float* nrm, float* expmd,
                       const int* bgcnt, int* prefix, int nblk) {
    int b = blockIdx.x;
    int t = threadIdx.x;
    if (t < KNUM) {
        float cm = fmaxf(cnt_k[b * KNUM + t], 1.0f);
        float sq = 0.f;
        for (int d = 0; d < FEATURE_DIM; ++d) {
            float m = (t == 0) ? 0.0f
                               : sums[(b * KNUM + t) * FEATURE_DIM + d] / cm;
            mean[(b * KNUM + t) * FEATURE_DIM + d] = m;
            sq += m * m;
        }
        nrm[b * KNUM + t]   = sqrtf(sq);
        expmd[b * KNUM + t] = expf(md[b * KNUM + t]);
    }
    if (t == 0) {
        int run = 0;
        for (int i = 0; i < nblk; ++i) {
            prefix[b * nblk + i] = run;
            run += bgcnt[b * nblk + i];
        }
    }
}

// ---------------- deterministic gather of first 100 background pixels -------
__global__ void k_bg(const float* __restrict__ emb, const int* __restrict__ inst,
                     const float* __restrict__ tmask, const int* __restrict__ prefix,
                     float* embbg, int N, int nblk) {
    int b   = blockIdx.x / nblk;
    int blk = blockIdx.x - b * nblk;
    int base = prefix[b * nblk + blk];
    if (base >= BG_SAMPLE) return;          // uniform exit
    const float* eb = emb + (size_t)b * FEATURE_DIM * N;
    const int*   ib = inst + (size_t)b * N;
    const float* tb = tmask + (size_t)b * N;
    __shared__ int s_wc[8];
    int t = threadIdx.x;
    int lane = t & 31, wave = t >> 5;
    int nw = blockDim.x >> 5;
    int beg = blk * CHUNK;
    int end = beg + CHUNK; if (end > N) end = N;
    int run = base;
    for (int start = beg; start < end; start += blockDim.x) {
        int i = start + t;
        bool pred = false;
        if (i < end) {
            int lab = ib[i];
            pred = (((tb[i] > 0.5f) ? lab : 0) == 0);
        }
        unsigned mask = (unsigned)__ballot(pred);   // wave32: low 32 bits
        int wcnt  = __popc(mask);
        int wrank = __popc(mask & ((1u << lane) - 1u));
        if (lane == 0) s_wc[wave] = wcnt;
        __syncthreads();
        int woff = 0, tot = 0;
        for (int w = 0; w < nw; ++w) { if (w < wave) woff += s_wc[w]; tot += s_wc[w]; }
        if (pred) {
            int rank = run + woff + wrank;
            if (rank < BG_SAMPLE) {
                float* o = embbg + ((size_t)b * BG_SAMPLE + rank) * FEATURE_DIM;
                o[0] = eb[i];
                o[1] = eb[(size_t)N + i];
                o[2] = eb[(size_t)2 * N + i];
                o[3] = eb[(size_t)3 * N + i];
            }
        }
        run += tot;
        __syncthreads();
        if (run >= BG_SAMPLE) return;       // uniform exit
    }
}

// ---------------- pass 2: l_agg value sums (needs means; emb hits L2) -------
__global__ void k_agg(const float* __restrict__ emb, const int* __restrict__ inst,
                      const float* __restrict__ tmask,
                      const float* __restrict__ mean, const float* __restrict__ expmd,
                      float* valsum, int N, int nblk) {
    int b   = blockIdx.x / nblk;
    int blk = blockIdx.x - b * nblk;
    __shared__ float s_mean[KNUM][FEATURE_DIM];
    __shared__ float s_scale[KNUM];
    __shared__ float s_val[KNUM];
    int t = threadIdx.x;
    if (t < KNUM) {
        for (int d = 0; d < FEATURE_DIM; ++d)
            s_mean[t][d] = mean[(b * KNUM + t) * FEATURE_DIM + d];
        s_scale[t] = expmd[b * KNUM + t];
        s_val[t] = 0.f;
    }
    __syncthreads();
    const float* eb = emb + (size_t)b * FEATURE_DIM * N;
    const int*   ib = inst + (size_t)b * N;
    const float* tb = tmask + (size_t)b * N;
    int beg = blk * CHUNK;
    int end = beg + CHUNK; if (end > N) end = N;
    for (int i = beg + t; i < end; i += blockDim.x) {
        int lab = ib[i];
        int l = (tb[i] > 0.5f) ? lab : 0;
        float d0 = eb[i] - s_mean[l][0];
        float d1 = eb[(size_t)N + i] - s_mean[l][1];
        float d2 = eb[(size_t)2 * N + i] - s_mean[l][2];
        float d3 = eb[(size_t)3 * N + i] - s_mean[l][3];
        float dist = sqrtf(d0 * d0 + d1 * d1 + d2 * d2 + d3 * d3);
        float u = fmaxf(s_scale[l] * dist - DELTA_V, 0.0f);
        atomicAdd(&s_val[l], logf(u * u + 1.0f));
    }
    __syncthreads();
    if (t < KNUM) atomicAdd(&valsum[b * KNUM + t], s_val[t]);
}

// ---------------- per-batch finalize: WMMA Gram of means + bg + reduce ------
__global__ void __launch_bounds__(32)
k_final(const float* __restrict__ mean, const float* __restrict__ nrm,
        const float* __restrict__ valsum, const float* __restrict__ cnt_all,
        const float* __restrict__ embbg, float* loss) {
    int b = blockIdx.x;
    int lane = threadIdx.x & 31;
    __shared__ float sm[KNUM][FEATURE_DIM];
    __shared__ float sn[KNUM];
    __shared__ float sbg[BG_SAMPLE][FEATURE_DIM];
    for (int d = 0; d < FEATURE_DIM; ++d)
        sm[lane][d] = mean[(b * KNUM + lane) * FEATURE_DIM + d];
    sn[lane] = nrm[b * KNUM + lane];
    for (int i = lane; i < BG_SAMPLE * FEATURE_DIM; i += 32)
        ((float*)sbg)[i] = embbg[(size_t)b * BG_SAMPLE * FEATURE_DIM + i];
    __syncthreads();

    const float coef = 1.0f - expf(-10.0f / (float)KNUM);

    // Gram G = M (32x4) * M^T (4x32) via four V_WMMA_F32_16X16X4_F32 tiles.
    // A layout (16x4 f32): lanes 0-15 hold M=lane, K={0,1}; lanes 16-31 K={2,3}.
    // For B = A^T of the same matrix the ISA B layout coincides with A's regs.
    int r  = lane & 15;
    int kb = (lane < 16) ? 0 : 2;
    v2f a0, a1;
    a0.x = sm[r][kb];       a0.y = sm[r][kb + 1];
    a1.x = sm[16 + r][kb];  a1.y = sm[16 + r][kb + 1];
    v8f g00 = {}, g01 = {}, g10 = {}, g11 = {};
    g00 = __builtin_amdgcn_wmma_f32_16x16x4_f32(false, a0, false, a0, (short)0, g00, false, false);
    g01 = __builtin_amdgcn_wmma_f32_16x16x4_f32(false, a0, false, a1, (short)0, g01, false, false);
    g10 = __builtin_amdgcn_wmma_f32_16x16x4_f32(false, a1, false, a0, (short)0, g10, false, false);
    g11 = __builtin_amdgcn_wmma_f32_16x16x4_f32(false, a1, false, a1, (short)0, g11, false, false);

    // pairwise term: pd^2 = |m_i|^2 + |m_j|^2 - 2 G_ij, mask diag & label 0
    float acc = 0.0f;
    int ncol = lane & 15;
    int mofs = (lane >= 16) ? 8 : 0;
    for (int e = 0; e < 8; ++e) {
        int M0 = e + mofs;
        float gv[4] = { g00[e], g01[e], g10[e], g11[e] };
        int Ms[4] = { M0, M0, M0 + 16, M0 + 16 };
        int Ns[4] = { ncol, ncol + 16, ncol, ncol + 16 };
        for (int tt = 0; tt < 4; ++tt) {
            int M = Ms[tt], Nc = Ns[tt];
            if (M != Nc && M != 0 && Nc != 0) {
                float pd2 = sn[M] * sn[M] + sn[Nc] * sn[Nc] - 2.0f * gv[tt];
                float pd  = sqrtf(fmaxf(pd2, 0.0f));
                float u   = fmaxf(2.0f * DELTA_D - coef * pd, 0.0f);
                acc += logf(u * u + 1.0f);
            }
        }
    }

    // background term: lane k handles label k vs 100 bg pixels
    float accbg = 0.0f;
    if (lane >= 1) {
        float m0 = sm[lane][0], m1 = sm[lane][1], m2 = sm[lane][2], m3 = sm[lane][3];
        for (int j = 0; j < BG_SAMPLE; ++j) {
            float d0 = sbg[j][0] - m0;
            float d1 = sbg[j][1] - m1;
            float d2 = sbg[j][2] - m2;
            float d3 = sbg[j][3] - m3;
            float dist = sqrtf(d0 * d0 + d1 * d1 + d2 * d2 + d3 * d3);
            float u = fmaxf(2.0f * DELTA_D - coef * dist, 0.0f);
            accbg += logf(u * u + 1.0f);
        }
    }
    const float denom = (float)((KNUM - 1) * (KNUM - 2) + (KNUM - 1));
    float part = (acc + accbg * (1.0f / (float)BG_SAMPLE)) / denom;

    // l_agg: mean over labels 1..K-1 of valsum/max(cnt,1)
    if (lane >= 1) {
        float c = fmaxf(cnt_all[b * KNUM + lane], 1.0f);
        part += (valsum[b * KNUM + lane] / c) / (float)(KNUM - 1);
    }
    // l_reg
    part += logf(sn[lane] + 1.0f) * (0.001f / (float)KNUM);

    for (int off = 16; off >= 1; off >>= 1)
        part += __shfl_xor(part, off, 32);
    if (lane == 0) loss[b] = part;
}

// ---------------- batch mean ----------------
__global__ void k_mean(const float* loss, float* out, int B) {
    if (threadIdx.x == 0 && blockIdx.x == 0) {
        float s = 0.0f;
        for (int b = 0; b < B; ++b) s += loss[b];
        out[0] = s / (float)B;   // LOSS_WEIGHT = 1
    }
}

extern "C" void kernel_launch(void* const* d_in, const int* in_sizes, int n_in,
                              void* d_out, int out_size, void* d_ws, size_t ws_size,
                              hipStream_t stream) {
    const float* emb   = (const float*)d_in[0];
    const int*   inst  = (const int*)d_in[1];
    const float* ker   = (const float*)d_in[2];
    const float* tmask = (const float*)d_in[3];
    /* d_in[4] = bboxes, unused by the reference loss */
    const float* md    = (const float*)d_in[5];

    int B = in_sizes[5] / KNUM;          // max_distances is [B, K]
    int N = in_sizes[1] / B;             // pixels per image
    int nblk = (N + CHUNK - 1) / CHUNK;

    float* ws      = (float*)d_ws;
    float* cnt_k   = ws;                                   // B*K
    float* sums    = cnt_k + B * KNUM;                     // B*K*4
    float* cnt_all = sums + B * KNUM * FEATURE_DIM;        // B*K
    float* valsum  = cnt_all + B * KNUM;                   // B*K
    float* embbg   = valsum + B * KNUM;                    // B*100*4
    int zcount = B * KNUM * 7 + B * BG_SAMPLE * FEATURE_DIM;
    float* mean    = embbg + B * BG_SAMPLE * FEATURE_DIM;  // B*K*4
    float* nrm     = mean + B * KNUM * FEATURE_DIM;        // B*K
    float* expmd   = nrm + B * KNUM;                       // B*K
    float* loss    = expmd + B * KNUM;                     // B
    int* bgcnt     = (int*)(loss + B);                     // B*nblk
    int* prefix    = bgcnt + B * nblk;                     // B*nblk

    k_zero<<<(zcount + 255) / 256, 256, 0, stream>>>(ws, zcount);
    k_stats<<<B * nblk, 256, 0, stream>>>(emb, inst, ker, tmask,
                                          cnt_k, sums, cnt_all, bgcnt, N, nblk);
    k_prep<<<B, 64, 0, stream>>>(cnt_k, sums, md, mean, nrm, expmd,
                                 bgcnt, prefix, nblk);
    k_bg<<<B * nblk, 256, 0, stream>>>(emb, inst, tmask, prefix, embbg, N, nblk);
    k_agg<<<B * nblk, 256, 0, stream>>>(emb, inst, tmask, mean, expmd,
                                        valsum, N, nblk);
    k_final<<<B, 32, 0, stream>>>(mean, nrm, valsum, cnt_all, embbg, loss);
    k_mean<<<1, 32, 0, stream>>>(loss, (float*)d_out, B);
}